// MHSA_42614665511380
// MI455X (gfx1250) — compile-verified
//
#include <hip/hip_runtime.h>
#include <hip/hip_bf16.h>

// MHSA (no-softmax linear attention) for gfx1250 / MI455X, wave32 WMMA fp32.
//
// Math factorization (attn has no softmax, so it is linear):
//   coarse_attn[h,k] = scale * dot(qsum[h], K[h,k,:]),  qsum[h] = sum_q Q[h,q,:]
//   output[h,q,:]    = scale * Q[h,q,:] @ (K^T V)[h]    (KtV is 64x64 per head)
// This removes the 8x4096x4096 attn tensor (512 MB, 34 GFLOP) and leaves the
// QKV GEMM (6.4 GFLOP) + 2x134 MMAC of tiny GEMMs: memory-roofline bound
// (~70 MB total traffic @ 23.3 TB/s, qkv intermediate L2-resident in 192 MB L2).
// fp32 V_WMMA_F32_16X16X4_F32 keeps full reference precision.
//
// Workspace layout (requires ~25.3 MB):
//   qkv  : 4096*1536 f32   (Q|K|V concatenated per token row)
//   KtV  : 8*64*64   f32   (atomic-accumulated partials)
//   qsum : 8*64      f32   (atomic-accumulated partials)

typedef __attribute__((ext_vector_type(2))) float v2f;
typedef __attribute__((ext_vector_type(4))) float v4f;
typedef __attribute__((ext_vector_type(8))) float v8f;

#define NTOK   4096
#define EMBED  512
#define QKVW   1536
#define HEADS  8
#define DH     64
#define SCALE  0.125f   // 64^-0.5

static __device__ __forceinline__ v8f wmma4(v2f a, v2f b, v8f c) {
    // D = A(16x4 f32) * B(4x16 f32) + C(16x16 f32)
    return __builtin_amdgcn_wmma_f32_16x16x4_f32(
        /*neg_a=*/false, a, /*neg_b=*/false, b,
        /*c_mod=*/(short)0, c, /*reuse_a=*/false, /*reuse_b=*/false);
}

// ---------------------------------------------------------------- zero scratch
__global__ void mhsa_zero(float* __restrict__ p, int n) {
    int i = blockIdx.x * blockDim.x + threadIdx.x;
    if (i < n) p[i] = 0.0f;
}

// ---------------------------------------------------------------- QKV GEMM
// qkv[n, c] = sum_e x[e, n] * W[e, c] + b[c]
// x is stored [E][N] (i.e. A-matrix transposed in memory -> contiguous in n,
// which is exactly the per-lane coalescing the A fragment wants).
// Wave tile: 16 tokens x 64 features. Workgroup: 4 waves stacked along tokens.
__global__ __launch_bounds__(128) void mhsa_qkv_gemm(
    const float* __restrict__ x, const float* __restrict__ W,
    const float* __restrict__ bias, float* __restrict__ qkv)
{
    const int lane = threadIdx.x & 31;
    const int wave = threadIdx.x >> 5;
    const int lo = lane & 15;        // A: row within tile / B,D: column
    const int hi = lane >> 4;        // lane-half selects K pair (0,1)/(2,3)
    const int m0 = blockIdx.x * 64 + wave * 16;  // token tile base
    const int n0 = blockIdx.y * 64;              // feature tile base

    v8f acc0 = {}, acc1 = {}, acc2 = {}, acc3 = {};
    const int m  = m0 + lo;          // A row this lane owns
    const int nb = n0 + lo;          // B column this lane owns

    #pragma unroll 4
    for (int k = 0; k < EMBED; k += 4) {
        const int ka = k + 2 * hi;
        v2f a;
        a.x = x[(size_t)ka * NTOK + m];
        a.y = x[(size_t)(ka + 1) * NTOK + m];
        const float* w0 = W + (size_t)ka * QKVW + nb;
        const float* w1 = W + (size_t)(ka + 1) * QKVW + nb;
        v2f b0 = {w0[0],  w1[0]};
        v2f b1 = {w0[16], w1[16]};
        v2f b2 = {w0[32], w1[32]};
        v2f b3 = {w0[48], w1[48]};
        acc0 = wmma4(a, b0, acc0);
        acc1 = wmma4(a, b1, acc1);
        acc2 = wmma4(a, b2, acc2);
        acc3 = wmma4(a, b3, acc3);
    }

    const float bi0 = bias[nb], bi1 = bias[nb + 16];
    const float bi2 = bias[nb + 32], bi3 = bias[nb + 48];
    #pragma unroll
    for (int r = 0; r < 8; ++r) {
        const int row = m0 + hi * 8 + r;             // C/D row mapping
        float* dst = qkv + (size_t)row * QKVW + n0 + lo;
        dst[0]  = acc0[r] + bi0;
        dst[16] = acc1[r] + bi1;
        dst[32] = acc2[r] + bi2;
        dst[48] = acc3[r] + bi3;
    }
}

// ---------------------------------------------------------------- qsum partial
// qsum[h, d] += sum over a 256-token chunk of Q[n, h*64+d]
__global__ __launch_bounds__(64) void mhsa_qsum(
    const float* __restrict__ qkv, float* __restrict__ qsum)
{
    const int h = blockIdx.x, chunk = blockIdx.y, d = threadIdx.x;
    const float* col = qkv + (size_t)chunk * 256 * QKVW + h * DH + d;
    float s = 0.0f;
    #pragma unroll 8
    for (int i = 0; i < 256; ++i) s += col[(size_t)i * QKVW];
    atomicAdd(&qsum[h * DH + d], s);
}

// ---------------------------------------------------------------- K^T V partial
// KtV[h, d1, d2] += sum over 256-token chunk of K[n,d1]*V[n,d2]
// A = K^T (rows d1, K-dim = tokens), B = V. Wave tile: 16 x 64 of the 64x64.
__global__ __launch_bounds__(128) void mhsa_ktv(
    const float* __restrict__ qkv, float* __restrict__ KtV)
{
    const int h = blockIdx.x, chunk = blockIdx.y;
    const int lane = threadIdx.x & 31, wave = threadIdx.x >> 5;
    const int lo = lane & 15, hi = lane >> 4;
    const int m0 = wave * 16;                       // d1 rows for this wave
    const float* Kb = qkv + EMBED + h * DH;         // K block base
    const float* Vb = qkv + 2 * EMBED + h * DH;     // V block base
    const int nbase = chunk * 256;

    v8f acc0 = {}, acc1 = {}, acc2 = {}, acc3 = {};
    #pragma unroll 4
    for (int kk = 0; kk < 256; kk += 4) {
        const int n = nbase + kk + 2 * hi;
        v2f a;                                       // A[d1][n] = K[n][d1]
        a.x = Kb[(size_t)n * QKVW + m0 + lo];
        a.y = Kb[(size_t)(n + 1) * QKVW + m0 + lo];
        const float* v0 = Vb + (size_t)n * QKVW + lo;
        const float* v1 = Vb + (size_t)(n + 1) * QKVW + lo;
        v2f b0 = {v0[0],  v1[0]};
        v2f b1 = {v0[16], v1[16]};
        v2f b2 = {v0[32], v1[32]};
        v2f b3 = {v0[48], v1[48]};
        acc0 = wmma4(a, b0, acc0);
        acc1 = wmma4(a, b1, acc1);
        acc2 = wmma4(a, b2, acc2);
        acc3 = wmma4(a, b3, acc3);
    }

    float* dst = KtV + (size_t)h * DH * DH;
    #pragma unroll
    for (int r = 0; r < 8; ++r) {
        const int row = m0 + hi * 8 + r;
        atomicAdd(&dst[row * DH + lo +  0], acc0[r]);
        atomicAdd(&dst[row * DH + lo + 16], acc1[r]);
        atomicAdd(&dst[row * DH + lo + 32], acc2[r]);
        atomicAdd(&dst[row * DH + lo + 48], acc3[r]);
    }
}

// ---------------------------------------------------------------- coarse_attn
// coarse[h, n] = SCALE * dot(qsum[h], K[n, h*64 : h*64+64])
__global__ __launch_bounds__(256) void mhsa_coarse(
    const float* __restrict__ qkv, const float* __restrict__ qsum,
    float* __restrict__ out)
{
    const int idx = blockIdx.x * blockDim.x + threadIdx.x;  // h*4096 + n
    const int h = idx >> 12, n = idx & (NTOK - 1);
    const float* Kr = qkv + (size_t)n * QKVW + EMBED + h * DH;  // 16B aligned
    const float* qs = qsum + h * DH;
    float s = 0.0f;
    #pragma unroll
    for (int d = 0; d < DH; d += 4) {
        const v4f kv = *(const v4f*)(Kr + d);
        const v4f qv = *(const v4f*)(qs + d);
        s += kv.x * qv.x + kv.y * qv.y + kv.z * qv.z + kv.w * qv.w;
    }
    out[idx] = s * SCALE;
}

// ---------------------------------------------------------------- output GEMM
// out[h, q, d] = SCALE * sum_d' Q[q, h*64+d'] * KtV[h, d', d]
// KtV tile (64x64 = 16 KB) staged in LDS once per workgroup; B frags read via
// ds_load. A frag per-lane loads are adjacent K pairs -> b64 global loads.
__global__ __launch_bounds__(128) void mhsa_out_gemm(
    const float* __restrict__ qkv, const float* __restrict__ KtV,
    float* __restrict__ out)
{
    __shared__ float lB[DH * DH];
    const int h = blockIdx.x, mt = blockIdx.y;

    {   // cooperative 16 KB LDS fill with b128 loads/stores
        const v4f* src = (const v4f*)(KtV + (size_t)h * DH * DH);
        v4f* dst = (v4f*)lB;
        for (int i = threadIdx.x; i < (DH * DH) / 4; i += 128) dst[i] = src[i];
    }
    __syncthreads();

    const int lane = threadIdx.x & 31, wave = threadIdx.x >> 5;
    const int lo = lane & 15, hi = lane >> 4;
    const int m0 = mt * 64 + wave * 16;
    const float* Qb = qkv + h * DH;
    const int m = m0 + lo;

    v8f acc0 = {}, acc1 = {}, acc2 = {}, acc3 = {};
    #pragma unroll
    for (int k = 0; k < DH; k += 4) {
        const int ka = k + 2 * hi;                        // even -> 8B aligned
        v2f a = *(const v2f*)(Qb + (size_t)m * QKVW + ka);
        const float* B0 = lB + ka * DH + lo;
        const float* B1 = lB + (ka + 1) * DH + lo;
        v2f b0 = {B0[0],  B1[0]};
        v2f b1 = {B0[16], B1[16]};
        v2f b2 = {B0[32], B1[32]};
        v2f b3 = {B0[48], B1[48]};
        acc0 = wmma4(a, b0, acc0);
        acc1 = wmma4(a, b1, acc1);
        acc2 = wmma4(a, b2, acc2);
        acc3 = wmma4(a, b3, acc3);
    }

    float* base = out + HEADS * NTOK + (size_t)h * NTOK * DH;  // after coarse
    #pragma unroll
    for (int r = 0; r < 8; ++r) {
        const int row = m0 + hi * 8 + r;
        float* o = base + (size_t)row * DH + lo;
        o[0]  = acc0[r] * SCALE;
        o[16] = acc1[r] * SCALE;
        o[32] = acc2[r] * SCALE;
        o[48] = acc3[r] * SCALE;
    }
}

// ---------------------------------------------------------------- launch
extern "C" void kernel_launch(void* const* d_in, const int* in_sizes, int n_in,
                              void* d_out, int out_size, void* d_ws, size_t ws_size,
                              hipStream_t stream) {
    const float* x    = (const float*)d_in[0];  // [512, 64*64] = [E][N]
    const float* W    = (const float*)d_in[1];  // [512, 1536]
    const float* bias = (const float*)d_in[2];  // [1536]
    float* out = (float*)d_out;                 // coarse (8*4096) ++ output (8*4096*64)

    float* qkv  = (float*)d_ws;                 // 4096*1536 f32
    float* KtV  = qkv + (size_t)NTOK * QKVW;    // 8*64*64 f32
    float* qsum = KtV + HEADS * DH * DH;        // 8*64 f32

    const int nzero = HEADS * DH * DH + HEADS * DH;   // KtV ++ qsum, contiguous
    mhsa_zero<<<(nzero + 255) / 256, 256, 0, stream>>>(KtV, nzero);

    mhsa_qkv_gemm<<<dim3(NTOK / 64, QKVW / 64), 128, 0, stream>>>(x, W, bias, qkv);
    mhsa_qsum<<<dim3(HEADS, NTOK / 256), 64, 0, stream>>>(qkv, qsum);
    mhsa_ktv<<<dim3(HEADS, NTOK / 256), 128, 0, stream>>>(qkv, KtV);
    mhsa_coarse<<<(HEADS * NTOK) / 256, 256, 0, stream>>>(qkv, qsum, out);
    mhsa_out_gemm<<<dim3(HEADS, NTOK / 64), 128, 0, stream>>>(qkv, KtV, out);
}